// myLoss_81810537054260
// MI455X (gfx1250) — compile-verified
//
#include <hip/hip_runtime.h>

typedef float v2f __attribute__((ext_vector_type(2)));
typedef float v8f __attribute__((ext_vector_type(8)));

#define BH 16
#define HH 768
#define WW 768
#define KK 512
#define WORDS_PER_ROW (WW / 32)                 // 24
#define WORDS_PER_BATCH (HH * WORDS_PER_ROW)    // 18432
#define MASK_WORDS (BH * WORDS_PER_BATCH)       // 294912 u32 = 1.15 MB
#define NUM_ROWBLOCKS (BH * HH)                 // 12288 partials

// ---------------------------------------------------------------------------
// Kernel 1: zero the gt bitmask region of the workspace
// ---------------------------------------------------------------------------
__global__ void zero_mask_kernel(unsigned int* __restrict__ mask) {
    int i = blockIdx.x * blockDim.x + threadIdx.x;
    if (i < MASK_WORDS) mask[i] = 0u;
}

// ---------------------------------------------------------------------------
// Kernel 2: scatter gt_cors -> bitmask  (gt_cors[b][k] = {x, y})
// ---------------------------------------------------------------------------
__global__ void scatter_kernel(const int* __restrict__ cors,
                               unsigned int* __restrict__ mask) {
    int t = blockIdx.x * blockDim.x + threadIdx.x;
    if (t >= BH * KK) return;
    int b = t / KK;
    int x = cors[2 * t + 0];
    int y = cors[2 * t + 1];
    int bitpos = y * WW + x;
    atomicOr(&mask[b * WORDS_PER_BATCH + (bitpos >> 5)], 1u << (bitpos & 31));
}

// ---------------------------------------------------------------------------
// Kernel 3: fused peak-mask + weighted-error pass.
// One workgroup per (batch, row); 192 active lanes x 4 cols (float4 loads).
// Writes one deterministic partial sum per block.
// ---------------------------------------------------------------------------
__global__ void loss_main_kernel(const float* __restrict__ pre,
                                 const float* __restrict__ gt,
                                 const unsigned int* __restrict__ mask,
                                 float* __restrict__ partials) {
    const int blk = blockIdx.x;            // b*768 + y
    const int b   = blk / HH;
    const int y   = blk % HH;
    const int tid = threadIdx.x;

    float v = 0.0f;
    if (tid < WW / 4) {
        const int x0 = tid * 4;
        const size_t rowbase = ((size_t)b * HH + (size_t)y) * WW;
        const float* rowc = pre + rowbase;
        const float* rowg = gt + rowbase;

        float4 c4 = *(const float4*)(rowc + x0);
        float4 g4 = *(const float4*)(rowg + x0);

        float4 u4 = make_float4(0.f, 0.f, 0.f, 0.f);   // zero padding
        float4 d4 = make_float4(0.f, 0.f, 0.f, 0.f);
        if (y > 0)      u4 = *(const float4*)(rowc - WW + x0);
        if (y < HH - 1) d4 = *(const float4*)(rowc + WW + x0);

        float lft = (x0 > 0)          ? rowc[x0 - 1] : 0.0f;
        float rgt = (x0 + 4 < WW)     ? rowc[x0 + 4] : 0.0f;

        unsigned int mw = mask[b * WORDS_PER_BATCH + y * WORDS_PER_ROW + (x0 >> 5)];

        float cc[4] = {c4.x, c4.y, c4.z, c4.w};
        float uu[4] = {u4.x, u4.y, u4.z, u4.w};
        float dd[4] = {d4.x, d4.y, d4.z, d4.w};
        float gg[4] = {g4.x, g4.y, g4.z, g4.w};
        float nl[4] = {lft,  c4.x, c4.y, c4.z};
        float nr[4] = {c4.y, c4.z, c4.w, rgt};

#pragma unroll
        for (int j = 0; j < 4; ++j) {
            bool peak = (cc[j] > nl[j]) && (cc[j] > nr[j]) &&
                        (cc[j] > uu[j]) && (cc[j] > dd[j]);
            bool gb   = ((mw >> ((x0 & 31) + j)) & 1u) != 0u;
            float e   = cc[j] - gg[j];
            e *= e;
            float w   = (gg[j] > 0.0f) ? 5.0f : 1.0f;  // 5*head + (1-head)
            float m1  = (peak != gb) ? 1.0f : 0.0f;    // xor mask
            v += e * (w + m1);
        }
    }

    __shared__ float s[256];
    s[tid] = v;
    __syncthreads();
#pragma unroll
    for (int st = 128; st > 0; st >>= 1) {
        if (tid < st) s[tid] += s[tid + st];
        __syncthreads();
    }
    if (tid == 0) partials[blk] = s[0];
}

// ---------------------------------------------------------------------------
// Kernel 4: WMMA-chained final reduction (single wave32).
// A = ones(16x4), B = 64 partials/step => D[m][n] = sum_k B[k][n] + C[m][n].
// All rows of D are identical; row 0 (VGPR0, lanes 0..15) holds the 16
// column-sums. Result is layout-agnostic (every loaded value enters exactly
// once) and exact f32.  12288 partials = 192 wmma steps, no tail.
// ---------------------------------------------------------------------------
__global__ void reduce_wmma_kernel(const float* __restrict__ partials,
                                   float* __restrict__ out) {
    const int lane = threadIdx.x;          // 0..31, EXEC all-ones (no divergence)

    v2f a;
    a[0] = 1.0f; a[1] = 1.0f;              // all-ones A matrix
    v8f c = {};                            // zero accumulator

    for (int it = 0; it < NUM_ROWBLOCKS / 64; ++it) {
        const float* p = partials + it * 64;
        v2f bv;
        bv[0] = p[lane];
        bv[1] = p[lane + 32];
        // 8 args: (neg_a, A, neg_b, B, c_mod, C, reuse_a, reuse_b)
        c = __builtin_amdgcn_wmma_f32_16x16x4_f32(
            false, a, false, bv, (short)0, c, false, false);
    }

    __shared__ float s[32];
    s[lane] = c[0];                        // VGPR0: lanes 0..15 = D[0][0..15]
    __syncthreads();
    if (lane == 0) {
        float t = 0.0f;
#pragma unroll
        for (int i = 0; i < 16; ++i) t += s[i];
        out[0] = t * (1.0f / 16.0f);       // mean over batch (l1 + l2)
    }
}

// ---------------------------------------------------------------------------
extern "C" void kernel_launch(void* const* d_in, const int* in_sizes, int n_in,
                              void* d_out, int out_size, void* d_ws, size_t ws_size,
                              hipStream_t stream) {
    const float* pre  = (const float*)d_in[0];
    const float* gt   = (const float*)d_in[1];
    const int*   cors = (const int*)d_in[2];
    float*       out  = (float*)d_out;

    unsigned int* mask     = (unsigned int*)d_ws;
    float*        partials = (float*)((char*)d_ws + (size_t)MASK_WORDS * 4);

    zero_mask_kernel<<<(MASK_WORDS + 255) / 256, 256, 0, stream>>>(mask);
    scatter_kernel<<<(BH * KK + 255) / 256, 256, 0, stream>>>(cors, mask);
    loss_main_kernel<<<NUM_ROWBLOCKS, 256, 0, stream>>>(pre, gt, mask, partials);
    reduce_wmma_kernel<<<1, 32, 0, stream>>>(partials, out);
}